// ApproxNDCGLoss_59871844106536
// MI455X (gfx1250) — compile-verified
//
#include <hip/hip_runtime.h>

// ----------------------------------------------------------------------------
// ApproxNDCG loss for [4096 x 8192] fp32.
// Per-row bitonic sorts in LDS (one row per 1024-thread / 32-wave workgroup),
// rows streamed into LDS with gfx1250 async global->LDS DMA (ASYNCcnt path).
// ----------------------------------------------------------------------------

#define C_DIM   8192
#define T_DIM   1024
#define EPT     (C_DIM / T_DIM)        // 8 elements per thread
#define VEC_IT  (C_DIM / (T_DIM * 4))  // 2 float4 chunks per thread
#define EPS_F   1e-8f

// ---- CDNA5 async global->LDS copy (gfx1250), guarded -----------------------
#if defined(__AMDGCN__)
#  if __has_builtin(__builtin_amdgcn_global_load_async_to_lds_b128) && \
      __has_builtin(__builtin_amdgcn_s_wait_asynccnt)
#    define USE_ASYNC_LDS 1
#  else
#    define USE_ASYNC_LDS 0
#    warning "gfx1250 async-to-LDS builtins NOT found; plain loads fallback"
#  endif
#else
#  define USE_ASYNC_LDS 0
#endif

#if USE_ASYNC_LDS
typedef int v4i_t __attribute__((vector_size(4 * sizeof(int))));
typedef __attribute__((address_space(1))) v4i_t* g_v4i_ptr;  // global
typedef __attribute__((address_space(3))) v4i_t* l_v4i_ptr;  // LDS
#endif

__device__ __forceinline__ void copy16_to_lds(const float* gsrc, float* ldst) {
#if USE_ASYNC_LDS
  // Generic 'global' pointers are identity-mapped in flat space; LDS flat
  // addresses carry the LDS byte offset in their low 32 bits (ISA 10.2).
  __builtin_amdgcn_global_load_async_to_lds_b128(
      (g_v4i_ptr)(unsigned long long)gsrc,
      (l_v4i_ptr)(unsigned)(unsigned long long)ldst,
      /*offset=*/0, /*cpol=*/0);
#else
  *(float4*)ldst = *(const float4*)gsrc;
#endif
}

__device__ __forceinline__ void wait_async_copies() {
#if USE_ASYNC_LDS
  __builtin_amdgcn_s_wait_asynccnt(0);
#endif
}

// ---- bitonic sort, descending, whole-block cooperative ---------------------
template <bool WITH_VAL>
__device__ __forceinline__ void bitonic_sort_desc(float* sk, float* sv,
                                                  unsigned tid) {
  for (unsigned k = 2; k <= C_DIM; k <<= 1) {
    for (unsigned j = k >> 1; j > 0; j >>= 1) {
      __syncthreads();
#pragma unroll
      for (unsigned p = 0; p < (C_DIM / 2) / T_DIM; ++p) {
        unsigned t   = tid + p * T_DIM;            // pair id 0..C/2-1
        unsigned i   = ((t & ~(j - 1u)) << 1) | (t & (j - 1u));
        unsigned ixj = i | j;
        bool descend = ((i & k) == 0);
        float a = sk[i];
        float b = sk[ixj];
        bool sw = descend ? (a < b) : (a > b);
        if (sw) {
          sk[i] = b;
          sk[ixj] = a;
          if (WITH_VAL) {
            float va = sv[i], vb = sv[ixj];
            sv[i] = vb;
            sv[ixj] = va;
          }
        }
      }
    }
  }
  __syncthreads();
}

// ---- per-row NDCG loss -----------------------------------------------------
__global__ __launch_bounds__(T_DIM) void ndcg_row_kernel(
    const float* __restrict__ logits, const float* __restrict__ targets,
    float* __restrict__ row_loss) {
  __shared__ float sk[C_DIM];   // 32 KB keys
  __shared__ float sv[C_DIM];   // 32 KB values / reduction scratch

  const unsigned tid = threadIdx.x;
  const unsigned row = blockIdx.x;
  const float* lrow = logits + (size_t)row * C_DIM;
  const float* trow = targets + (size_t)row * C_DIM;

  // Stream logits->sk, targets->sv via async DMA (4 x b128 per thread).
#pragma unroll
  for (unsigned p = 0; p < VEC_IT; ++p) {
    unsigned c = (tid + p * T_DIM) * 4;
    copy16_to_lds(lrow + c, &sk[c]);
    copy16_to_lds(trow + c, &sv[c]);
  }
  wait_async_copies();
  __syncthreads();

  // Phase 1: sort (logit, target) pairs by logit descending -> pred DCG.
  bitonic_sort_desc<true>(sk, sv, tid);

  float partial = 0.0f;
#pragma unroll
  for (unsigned p = 0; p < EPT; ++p) {
    unsigned i = tid + p * T_DIM;
    partial += sv[i] * (1.0f / __log2f((float)(i + 2)));
  }

  // Overlap: prefetch targets into sk for phase 2 while reducing in sv.
#pragma unroll
  for (unsigned p = 0; p < VEC_IT; ++p) {
    unsigned c = (tid + p * T_DIM) * 4;
    copy16_to_lds(trow + c, &sk[c]);
  }

  __syncthreads();              // all reads of sv done
  sv[tid] = partial;
  for (unsigned s = T_DIM / 2; s > 0; s >>= 1) {
    __syncthreads();
    if (tid < s) sv[tid] += sv[tid + s];
  }
  __syncthreads();
  float pred_dcg = sv[0];

  wait_async_copies();          // each wave drains its own ASYNCcnt
  __syncthreads();              // then all waves' data visible

  // Phase 2: sort targets descending (keys only) -> ideal DCG.
  bitonic_sort_desc<false>(sk, sv, tid);

  float partial2 = 0.0f;
#pragma unroll
  for (unsigned p = 0; p < EPT; ++p) {
    unsigned i = tid + p * T_DIM;
    partial2 += sk[i] * (1.0f / __log2f((float)(i + 2)));
  }
  __syncthreads();
  sv[tid] = partial2;
  for (unsigned s = T_DIM / 2; s > 0; s >>= 1) {
    __syncthreads();
    if (tid < s) sv[tid] += sv[tid + s];
  }
  __syncthreads();
  if (tid == 0) {
    float ideal_dcg = sv[0];
    row_loss[row] = 1.0f - pred_dcg / (ideal_dcg + EPS_F);
  }
}

// ---- deterministic mean over rows ------------------------------------------
__global__ __launch_bounds__(1024) void reduce_mean_kernel(
    const float* __restrict__ row_loss, float* __restrict__ out, int nrows) {
  __shared__ float s[1024];
  unsigned tid = threadIdx.x;
  float acc = 0.0f;
  for (int i = (int)tid; i < nrows; i += 1024) acc += row_loss[i];
  s[tid] = acc;
  for (unsigned st = 512; st > 0; st >>= 1) {
    __syncthreads();
    if (tid < st) s[tid] += s[tid + st];
  }
  __syncthreads();
  if (tid == 0) out[0] = s[0] / (float)nrows;
}

// ----------------------------------------------------------------------------
extern "C" void kernel_launch(void* const* d_in, const int* in_sizes, int n_in,
                              void* d_out, int out_size, void* d_ws,
                              size_t ws_size, hipStream_t stream) {
  (void)n_in; (void)out_size; (void)ws_size;
  const float* logits  = (const float*)d_in[0];
  const float* targets = (const float*)d_in[1];
  float* out = (float*)d_out;
  float* ws  = (float*)d_ws;   // B floats of per-row loss

  const int B = in_sizes[0] / C_DIM;   // 4096

  ndcg_row_kernel<<<dim3(B), dim3(T_DIM), 0, stream>>>(logits, targets, ws);
  reduce_mean_kernel<<<dim3(1), dim3(1024), 0, stream>>>(ws, out, B);
}